// BlockLinear_44427141710514
// MI455X (gfx1250) — compile-verified
//
#include <hip/hip_runtime.h>

// Block-sparse linear: async gather -> per-block GEMM (f32 WMMA) -> native f32
// atomic scatter-add + bias.  MI455X / gfx1250, wave32.

typedef __attribute__((ext_vector_type(2))) float v2f;
typedef __attribute__((ext_vector_type(8))) float v8f;

#define IN_F   4096
#define OUT_F  4096
#define NB     64
#define BIN    512
#define BOUT   512
#define TM     128          // tokens per workgroup (8 waves x 16 tokens)
#define XG_STRIDE 514       // 512 + 2 pad: float2 LDS reads 8B-aligned, banks spread
#define W_STRIDE  68        // 64 + 4 pad: b128 async LDS writes 16B-aligned,
                            // A-frag b64 reads conflict-free (4 banks/lane step)
#define WCHUNK (64 * W_STRIDE)

// low 32 bits of a generic shared-memory pointer == LDS byte offset
__device__ __forceinline__ uint32_t lds_off(const void* p) {
    return (uint32_t)(uintptr_t)p;
}
// CDNA5 async global->LDS copies (ASYNCcnt-tracked, no VGPR round-trip)
__device__ __forceinline__ void async_load_b32(uint32_t lds, const float* g) {
    asm volatile("global_load_async_to_lds_b32 %0, %1, off" :: "v"(lds), "v"(g) : "memory");
}
__device__ __forceinline__ void async_load_b128(uint32_t lds, const float* g) {
    asm volatile("global_load_async_to_lds_b128 %0, %1, off" :: "v"(lds), "v"(g) : "memory");
}
__device__ __forceinline__ void wait_async0() {
    asm volatile("s_wait_asynccnt 0" ::: "memory");
}

// ---------------------------------------------------------------------------
// Kernel 1: out[t, f] = bias[f]  (d_out is poisoned by the harness)
// ---------------------------------------------------------------------------
__global__ __launch_bounds__(256, 1)
void bias_init_kernel(const float* __restrict__ bias, float* __restrict__ out, int n4)
{
    int i = blockIdx.x * 256 + threadIdx.x;
    if (i < n4) {
        int f4 = i & (OUT_F / 4 - 1);
        ((float4*)out)[i] = ((const float4*)bias)[f4];
    }
}

// ---------------------------------------------------------------------------
// Kernel 2: per-(block, token-tile) gathered GEMM with V_WMMA_F32_16X16X4_F32.
//   A = W[o, k]   (16x4; lane l<16: row o=l, K={k,k+1}; lane>=16: K={k+2,k+3})
//   B = x^T[k, t] (4x16; lane l<16: col t=l, rows K={k,k+1}; lane>=16: K={k+2,k+3})
//   D = [o x t]   (VGPR v: lanes 0-15 row o=v, lanes 16-31 row o=v+8)
// W chunks are double-buffered via global_load_async_to_lds_b128.
// ---------------------------------------------------------------------------
__global__ __launch_bounds__(256, 1)
void block_linear_kernel(const float* __restrict__ x,
                         const float* __restrict__ w,
                         const int*   __restrict__ in_idx,
                         const int*   __restrict__ out_idx,
                         float*       __restrict__ out)
{
    __shared__ __align__(16) float xg[TM * XG_STRIDE];  // gathered x [128 tok][512 k]
    __shared__ __align__(16) float wl[2 * WCHUNK];      // double-buffered W [64 o][64 k]
    __shared__ int iin[BIN];
    __shared__ int iout[BOUT];

    const int tid   = threadIdx.x;
    const int n     = blockIdx.y;
    const int tBase = blockIdx.x * TM;
    const int lane  = tid & 31;
    const int wave  = tid >> 5;
    const int l16   = lane & 15;
    const int half  = lane >> 4;            // 0: K={0,1}, 1: K={2,3}

    // ---- stage indices ----
    for (int i = tid; i < BIN;  i += 256) iin[i]  = in_idx [n * BIN  + i];
    for (int i = tid; i < BOUT; i += 256) iout[i] = out_idx[n * BOUT + i];
    __syncthreads();

    // ---- async gather of the x tile into LDS (once per workgroup) ----
    const uint32_t xg_base = lds_off(xg);
    for (int i = tid; i < TM * BIN; i += 256) {
        int t = i >> 9;            // / 512
        int k = i & 511;
        async_load_b32(xg_base + (uint32_t)(t * XG_STRIDE + k) * 4u,
                       x + (size_t)(tBase + t) * IN_F + iin[k]);
    }

    const uint32_t wl_base = lds_off(wl);
    const int      tloc    = wave * 16;
    const float*   wblk    = w + (size_t)n * BOUT * BIN;

    for (int o0 = 0; o0 < BOUT; o0 += 64) {
        v8f d0 = {}, d1 = {}, d2 = {}, d3 = {};

        // prefetch first W chunk (kc=0) into buffer 0 (one b128 per thread x4)
        #pragma unroll
        for (int r = 0; r < 4; ++r) {
            int i4 = (tid + r * 256) * 4;
            int ol = i4 >> 6, kl = i4 & 63;
            async_load_b128(wl_base + (uint32_t)(ol * W_STRIDE + kl) * 4u,
                            wblk + (size_t)(o0 + ol) * BIN + kl);
        }

        for (int kc = 0; kc < BIN; kc += 64) {
            const int cur = (kc >> 6) & 1;
            wait_async0();          // my async writes (and gather) have landed
            __syncthreads();        // everyone's async writes visible

            if (kc + 64 < BIN) {    // issue next chunk into the other buffer
                const int nxt = cur ^ 1;
                #pragma unroll
                for (int r = 0; r < 4; ++r) {
                    int i4 = (tid + r * 256) * 4;
                    int ol = i4 >> 6, kl = i4 & 63;
                    async_load_b128(wl_base + (uint32_t)(nxt * WCHUNK + ol * W_STRIDE + kl) * 4u,
                                    wblk + (size_t)(o0 + ol) * BIN + (kc + 64) + kl);
                }
            }

            const float* wb = wl + cur * WCHUNK;
            #pragma unroll
            for (int kk = 0; kk < 64; kk += 4) {
                v2f b  = *(const v2f*)&xg[(tloc + l16) * XG_STRIDE + kc + kk + 2 * half];
                v2f a0 = *(const v2f*)&wb[( 0 + l16) * W_STRIDE + kk + 2 * half];
                v2f a1 = *(const v2f*)&wb[(16 + l16) * W_STRIDE + kk + 2 * half];
                v2f a2 = *(const v2f*)&wb[(32 + l16) * W_STRIDE + kk + 2 * half];
                v2f a3 = *(const v2f*)&wb[(48 + l16) * W_STRIDE + kk + 2 * half];

                d0 = __builtin_amdgcn_wmma_f32_16x16x4_f32(false, a0, false, b, (short)0, d0, false, false);
                d1 = __builtin_amdgcn_wmma_f32_16x16x4_f32(false, a1, false, b, (short)0, d1, false, false);
                d2 = __builtin_amdgcn_wmma_f32_16x16x4_f32(false, a2, false, b, (short)0, d2, false, false);
                d3 = __builtin_amdgcn_wmma_f32_16x16x4_f32(false, a3, false, b, (short)0, d3, false, false);
            }
        }

        // ---- scatter-add this o-chunk with native f32 atomics ----
        const size_t rowBase = (size_t)(tBase + tloc + l16) * OUT_F;
        #pragma unroll
        for (int v = 0; v < 8; ++v) {
            unsafeAtomicAdd(&out[rowBase + iout[o0 +  0 + v + 8 * half]], d0[v]);
            unsafeAtomicAdd(&out[rowBase + iout[o0 + 16 + v + 8 * half]], d1[v]);
            unsafeAtomicAdd(&out[rowBase + iout[o0 + 32 + v + 8 * half]], d2[v]);
            unsafeAtomicAdd(&out[rowBase + iout[o0 + 48 + v + 8 * half]], d3[v]);
        }
    }
}

// ---------------------------------------------------------------------------
extern "C" void kernel_launch(void* const* d_in, const int* in_sizes, int n_in,
                              void* d_out, int out_size, void* d_ws, size_t ws_size,
                              hipStream_t stream)
{
    const float* x    = (const float*)d_in[0];
    const float* w    = (const float*)d_in[1];
    const float* bias = (const float*)d_in[2];
    const int*   iin  = (const int*)  d_in[3];
    const int*   iout = (const int*)  d_in[4];
    float*       out  = (float*)d_out;

    const int totalTokens = 2 * 2048;                 // B * S
    const int n4 = totalTokens * OUT_F / 4;

    bias_init_kernel<<<(n4 + 255) / 256, 256, 0, stream>>>(bias, out, n4);

    dim3 grid(totalTokens / TM, NB);                  // 32 token tiles x 64 blocks
    block_linear_kernel<<<grid, 256, 0, stream>>>(x, w, iin, iout, out);
}